// GraphClassification_31293131719247
// MI455X (gfx1250) — compile-verified
//
#include <hip/hip_runtime.h>
#include <hip/hip_bf16.h>

// Problem sizes (fixed by the reference)
#define N_NODES  100000
#define E_EDGES  1600000
#define G_GRAPHS 1000
#define F_IN     16
#define D_DIM    128
#define C_CENT   100
#define C_PAD    112      // 7 tiles of 16
#define K_CLS    2
#define L_LAYERS 3

typedef __attribute__((ext_vector_type(2))) float v2f;
typedef __attribute__((ext_vector_type(8))) float v8f;

// Hardware fp32 atomic add (global_atomic_add_f32); targets L2 (arrays fit in 192MB L2).
__device__ __forceinline__ void atomAddF(float* p, float v) {
    unsafeAtomicAdd(p, v);
}

// ---------------------------------------------------------------------------
// Utility kernels
// ---------------------------------------------------------------------------
__global__ void fill_kernel(float* __restrict__ p, int n, float v) {
    int i = blockIdx.x * blockDim.x + threadIdx.x;
    int stride = gridDim.x * blockDim.x;
    for (; i < n; i += stride) p[i] = v;
}

__global__ void deg_edges_kernel(const int* __restrict__ dst, float* __restrict__ deg) {
    int e = blockIdx.x * blockDim.x + threadIdx.x;
    if (e < E_EDGES) atomAddF(&deg[dst[e]], 1.0f);
}

__global__ void deg_fin_kernel(float* __restrict__ deg) {
    int n = blockIdx.x * blockDim.x + threadIdx.x;
    if (n < N_NODES) deg[n] = rsqrtf(fmaxf(deg[n], 1.0f));
}

// ---------------------------------------------------------------------------
// Embedding: H[n][d] = sum_f X[n][f] * Wemb[d][f]   (16 nodes / block)
// ---------------------------------------------------------------------------
__global__ __launch_bounds__(256)
void embed_kernel(const float* __restrict__ X, const float* __restrict__ Wemb,
                  float* __restrict__ H) {
    __shared__ float sW[D_DIM * F_IN];  // 8 KB
    __shared__ float sX[16 * F_IN];     // 1 KB
    const int tid = threadIdx.x;
    const int n0  = blockIdx.x * 16;
    for (int i = tid; i < D_DIM * F_IN; i += 256) sW[i] = Wemb[i];
    for (int i = tid; i < 16 * F_IN;   i += 256) sX[i] = X[(size_t)n0 * F_IN + i];
    __syncthreads();
#pragma unroll
    for (int i = 0; i < 8; ++i) {
        int o  = tid + i * 256;      // 0..2047
        int nl = o >> 7;
        int d  = o & 127;
        float acc = 0.f;
#pragma unroll
        for (int f = 0; f < F_IN; ++f) acc += sX[nl * F_IN + f] * sW[d * F_IN + f];
        H[(size_t)(n0 + nl) * D_DIM + d] = acc;
    }
}

// ---------------------------------------------------------------------------
// M = (relu?)(H) @ W^T   with H:[N,128], W:[128,128]  (fp32 WMMA 16x16x4)
// Block: 256 threads = 8 waves; wave w owns output columns [16w,16w+16);
// block b owns rows [16b, 16b+16). W staged in 64 KB LDS.
// ---------------------------------------------------------------------------
template <bool RELU>
__global__ __launch_bounds__(256)
void gemm128_kernel(const float* __restrict__ H, const float* __restrict__ W,
                    float* __restrict__ M) {
    __shared__ float sW[D_DIM * D_DIM];  // 64 KB: W[j][k]
    const int tid = threadIdx.x;
    {
        const float4* W4  = (const float4*)W;
        float4*       sW4 = (float4*)sW;
#pragma unroll
        for (int i = 0; i < (D_DIM * D_DIM / 4) / 256; ++i)  // 16 iters
            sW4[tid + i * 256] = W4[tid + i * 256];
    }
    __syncthreads();

    const int wave = tid >> 5;       // 0..7
    const int lane = tid & 31;
    const int half = lane >> 4;      // 0/1
    const int l16  = lane & 15;
    const int row  = blockIdx.x * 16 + l16;
    const int j0   = wave * 16;

    const float2* Hrow = (const float2*)(H + (size_t)row * D_DIM);

    v8f acc = {0.f, 0.f, 0.f, 0.f, 0.f, 0.f, 0.f, 0.f};
#pragma unroll 8
    for (int k0 = 0; k0 < D_DIM; k0 += 4) {
        const int kk = k0 + 2 * half;
        float2 av = Hrow[kk >> 1];                 // H[row][kk], H[row][kk+1]
        if (RELU) { av.x = fmaxf(av.x, 0.f); av.y = fmaxf(av.y, 0.f); }
        v2f a; a.x = av.x; a.y = av.y;
        v2f b;                                      // B[k][j] = W[j][k]
        b.x = sW[(j0 + l16) * D_DIM + kk];
        b.y = sW[(j0 + l16) * D_DIM + kk + 1];
        acc = __builtin_amdgcn_wmma_f32_16x16x4_f32(false, a, false, b,
                                                    (short)0, acc, false, false);
    }

    float* out = M + (size_t)(blockIdx.x * 16) * D_DIM;
#pragma unroll
    for (int r = 0; r < 8; ++r) {
        int orow = r + half * 8;                    // D-matrix VGPR layout
        out[(size_t)orow * D_DIM + j0 + l16] = acc[r];
    }
}

// ---------------------------------------------------------------------------
// Aggregation: out[n] = M[n]*dis[n]^2 (self-loop, also serves as init),
// then out[dst] += M[src]*dis[src]*dis[dst] via fp32 hardware atomics.
// 32 threads per row (float4 each) -> fully coalesced 128-float rows.
// ---------------------------------------------------------------------------
__global__ __launch_bounds__(256)
void selfloop_init_kernel(const float* __restrict__ M, const float* __restrict__ dis,
                          float* __restrict__ Hout) {
    int idx = blockIdx.x * blockDim.x + threadIdx.x;   // N*32 threads
    int n = idx >> 5;
    int c = (idx & 31) << 2;
    if (n >= N_NODES) return;
    float d = dis[n];
    float co = d * d;
    const float4 v = *(const float4*)(M + (size_t)n * D_DIM + c);
    float4 o; o.x = v.x * co; o.y = v.y * co; o.z = v.z * co; o.w = v.w * co;
    *(float4*)(Hout + (size_t)n * D_DIM + c) = o;
}

__global__ __launch_bounds__(256)
void edge_scatter_kernel(const int* __restrict__ src, const int* __restrict__ dst,
                         const float* __restrict__ dis, const float* __restrict__ M,
                         float* __restrict__ Hout) {
    int idx = blockIdx.x * blockDim.x + threadIdx.x;   // E*32 threads
    int e = idx >> 5;
    int c = (idx & 31) << 2;
    if (e >= E_EDGES) return;
    int s = src[e], t = dst[e];
    float co = dis[s] * dis[t];
    const float4 v = *(const float4*)(M + (size_t)s * D_DIM + c);
    float* o = Hout + (size_t)t * D_DIM + c;
    atomAddF(o + 0, v.x * co);
    atomAddF(o + 1, v.y * co);
    atomAddF(o + 2, v.z * co);
    atomAddF(o + 3, v.w * co);
}

// ---------------------------------------------------------------------------
// Final relu (in place) + squared row norm
// ---------------------------------------------------------------------------
__global__ __launch_bounds__(128)
void relu_norm_kernel(float* __restrict__ H, float* __restrict__ x2) {
    __shared__ float red[128];
    const int n = blockIdx.x;
    float v = H[(size_t)n * D_DIM + threadIdx.x];
    v = fmaxf(v, 0.f);
    H[(size_t)n * D_DIM + threadIdx.x] = v;
    red[threadIdx.x] = v * v;
    __syncthreads();
    for (int s = 64; s > 0; s >>= 1) {
        if (threadIdx.x < s) red[threadIdx.x] += red[threadIdx.x + s];
        __syncthreads();
    }
    if (threadIdx.x == 0) x2[n] = red[0];
}

__global__ void cnt_kernel(const int* __restrict__ batch, float* __restrict__ cnt) {
    int n = blockIdx.x * blockDim.x + threadIdx.x;
    if (n < N_NODES) atomAddF(&cnt[batch[n]], 1.0f);
}

// ---------------------------------------------------------------------------
// Distance + pooled accumulation.
// dist[n][c] = sqrt(max(x2[n]+c2[c]-2*h[n].cent[c],0)+1e-12); atomic-add into
// sums[batch[n]][c]. Centroids padded to 112 cols (7 tiles), staged in LDS.
// Block: 224 threads = 7 waves; wave w owns centroid tile w; K=128 via f32 WMMA.
// ---------------------------------------------------------------------------
__global__ __launch_bounds__(224)
void dist_pool_kernel(const float* __restrict__ H, const float* __restrict__ cent,
                      const float* __restrict__ x2, const int* __restrict__ batch,
                      float* __restrict__ sums) {
    __shared__ float sC[C_PAD * D_DIM];  // 57344 B, padded rows = 0
    __shared__ float sc2[C_PAD];
    const int tid = threadIdx.x;
    for (int i = tid; i < C_PAD * D_DIM; i += 224) {
        int r = i >> 7, k = i & 127;
        sC[i] = (r < C_CENT) ? cent[(size_t)r * D_DIM + k] : 0.f;
    }
    __syncthreads();
    if (tid < C_PAD) {
        float s = 0.f;
        for (int k = 0; k < D_DIM; ++k) { float v = sC[tid * D_DIM + k]; s += v * v; }
        sc2[tid] = s;
    }
    __syncthreads();

    const int wave = tid >> 5;       // 0..6
    const int lane = tid & 31;
    const int half = lane >> 4;
    const int l16  = lane & 15;
    const int row  = blockIdx.x * 16 + l16;
    const int j0   = wave * 16;

    const float2* Hrow = (const float2*)(H + (size_t)row * D_DIM);

    v8f acc = {0.f, 0.f, 0.f, 0.f, 0.f, 0.f, 0.f, 0.f};
#pragma unroll 8
    for (int k0 = 0; k0 < D_DIM; k0 += 4) {
        const int kk = k0 + 2 * half;
        float2 av = Hrow[kk >> 1];
        v2f a; a.x = av.x; a.y = av.y;
        v2f b;                                      // B[k][j] = cent[j0+j][k]
        b.x = sC[(j0 + l16) * D_DIM + kk];
        b.y = sC[(j0 + l16) * D_DIM + kk + 1];
        acc = __builtin_amdgcn_wmma_f32_16x16x4_f32(false, a, false, b,
                                                    (short)0, acc, false, false);
    }

    const int col = j0 + l16;
#pragma unroll
    for (int r = 0; r < 8; ++r) {
        int node = blockIdx.x * 16 + r + half * 8;
        float d2 = x2[node] + sc2[col] - 2.f * acc[r];
        float d  = sqrtf(fmaxf(d2, 0.f) + 1e-12f);
        if (col < C_CENT)
            atomAddF(&sums[(size_t)batch[node] * C_CENT + col], d);
    }
}

// ---------------------------------------------------------------------------
// Head: out[g][k] = sum_c (sums[g][c]/max(cnt[g],1)) * Wout[k][c] + bout[k]
// ---------------------------------------------------------------------------
__global__ void final_kernel(const float* __restrict__ sums, const float* __restrict__ cnt,
                             const float* __restrict__ Wout, const float* __restrict__ bout,
                             float* __restrict__ out) {
    int idx = blockIdx.x * blockDim.x + threadIdx.x;
    if (idx >= G_GRAPHS * K_CLS) return;
    int g = idx / K_CLS, k = idx % K_CLS;
    float inv = 1.f / fmaxf(cnt[g], 1.f);
    float acc = bout[k];
    for (int c = 0; c < C_CENT; ++c)
        acc += (sums[(size_t)g * C_CENT + c] * inv) * Wout[k * C_CENT + c];
    out[idx] = acc;
}

// ---------------------------------------------------------------------------
extern "C" void kernel_launch(void* const* d_in, const int* in_sizes, int n_in,
                              void* d_out, int out_size, void* d_ws, size_t ws_size,
                              hipStream_t stream) {
    const float* x     = (const float*)d_in[0];
    const int*   eidx  = (const int*)  d_in[1];
    const int*   batch = (const int*)  d_in[2];
    const float* Wemb  = (const float*)d_in[3];
    const float* Wconv = (const float*)d_in[4];
    const float* cent  = (const float*)d_in[5];
    const float* Wout  = (const float*)d_in[6];
    const float* bout  = (const float*)d_in[7];
    float* out = (float*)d_out;

    // workspace layout (floats): two N*D ping-pong buffers + small arrays (~104 MB)
    float* ws   = (float*)d_ws;
    float* hA   = ws;                                    // [N,128] h
    float* hM   = hA + (size_t)N_NODES * D_DIM;          // [N,128] m = h@W^T
    float* dis  = hM + (size_t)N_NODES * D_DIM;          // [N] deg^{-1/2}
    float* x2   = dis + N_NODES;                         // [N] |h|^2
    float* sums = x2 + N_NODES;                          // [G,100]
    float* cnt  = sums + (size_t)G_GRAPHS * C_CENT;      // [G]

    const int* src = eidx;
    const int* dst = eidx + E_EDGES;

    // 1) degrees (self-loop contributes the initial 1.0)
    fill_kernel<<<512, 256, 0, stream>>>(dis, N_NODES, 1.0f);
    deg_edges_kernel<<<(E_EDGES + 255) / 256, 256, 0, stream>>>(dst, dis);
    deg_fin_kernel<<<(N_NODES + 255) / 256, 256, 0, stream>>>(dis);

    // 2) embedding
    embed_kernel<<<N_NODES / 16, 256, 0, stream>>>(x, Wemb, hA);

    // 3) three GCN layers: GEMM (WMMA) -> self-loop init -> edge scatter
    for (int l = 0; l < L_LAYERS; ++l) {
        const float* Wl = Wconv + (size_t)l * D_DIM * D_DIM;
        if (l == 0)
            gemm128_kernel<false><<<N_NODES / 16, 256, 0, stream>>>(hA, Wl, hM);
        else
            gemm128_kernel<true ><<<N_NODES / 16, 256, 0, stream>>>(hA, Wl, hM);
        selfloop_init_kernel<<<(N_NODES * 32) / 256, 256, 0, stream>>>(hM, dis, hA);
        edge_scatter_kernel<<<(E_EDGES * 32) / 256, 256, 0, stream>>>(src, dst, dis, hM, hA);
    }

    // 4) final relu + row norms
    relu_norm_kernel<<<N_NODES, 128, 0, stream>>>(hA, x2);

    // 5) pooling accumulators (sums and cnt are contiguous)
    fill_kernel<<<256, 256, 0, stream>>>(sums, G_GRAPHS * C_CENT + G_GRAPHS, 0.0f);
    cnt_kernel<<<(N_NODES + 255) / 256, 256, 0, stream>>>(batch, cnt);

    // 6) centroid distances (WMMA) + per-graph accumulation
    dist_pool_kernel<<<N_NODES / 16, 224, 0, stream>>>(hA, cent, x2, batch, sums);

    // 7) classification head
    final_kernel<<<(G_GRAPHS * K_CLS + 255) / 256, 256, 0, stream>>>(sums, cnt, Wout, bout, out);
}